// SymmetricCrossEntropy_11751030522421
// MI455X (gfx1250) — compile-verified
//
#include <hip/hip_runtime.h>
#include <cstdint>
#include <cstddef>

// Symmetric cross entropy:  out = CE + RCE  (scalar, fp32)
//   CE  = -mean(log_softmax(logits)[n, t])
//   RCE = -mean(LOG_EPS * (sum(clip(p,1e-7,1)) - clip(p_t,1e-7,1)))
// Bandwidth-bound: 262 MB streamed once via GLOBAL_LOAD_ASYNC_TO_LDS_B128.

#define NUM_C 1000
#define C_PAD 1024          // 4 KB LDS buffer per wave (pad with -inf)
#define ROWS_PER_BLOCK 8    // 8 wave32s per 256-thread block, 1 row per wave
#define NEG_LOG_EPS 9.210340371976184f  // -log(1e-4)

__global__ void __launch_bounds__(256)
sce_row_kernel(const float* __restrict__ logits,
               const int* __restrict__ targets,
               float* __restrict__ partials,
               int n_rows) {
  __shared__ float smem[ROWS_PER_BLOCK * C_PAD];
  __shared__ float sterm[ROWS_PER_BLOCK];

  const int lane = threadIdx.x & 31;
  const int wave = threadIdx.x >> 5;
  const int row  = blockIdx.x * ROWS_PER_BLOCK + wave;
  float* buf = smem + wave * C_PAD;

  float term = 0.0f;
  if (row < n_rows) {
    // Pad LDS tail [1000,1024) with -inf: max/sum-exp passes need no masking.
    if (lane < (C_PAD - NUM_C)) buf[NUM_C + lane] = -__builtin_inff();

    // ---- async copy: 4000 B global -> LDS, 16 B per lane, 512 B per issue
    const uint64_t grow = (uint64_t)(uintptr_t)(logits + (size_t)row * NUM_C);
    const uint32_t lrow = (uint32_t)(uintptr_t)buf;  // low 32 bits = LDS addr
#pragma unroll
    for (int k = 0; k < 8; ++k) {
      const int byteoff = (k * 32 + lane) * 16;
      if (byteoff < NUM_C * 4) {   // tail issue: 26 of 32 lanes active
        uint32_t laddr = lrow + (uint32_t)byteoff;
        uint64_t gaddr = grow + (uint64_t)byteoff;
        asm volatile("global_load_async_to_lds_b128 %0, %1, off"
                     :: "v"(laddr), "v"(gaddr)
                     : "memory");
      }
    }
    asm volatile("s_wait_asynccnt 0x0" ::: "memory");

    // ---- pull the whole row into VGPRs (8 x float4 per lane)
    const float4* buf4 = (const float4*)buf;
    float4 r[8];
#pragma unroll
    for (int k = 0; k < 8; ++k) r[k] = buf4[k * 32 + lane];

    // ---- pass 1: row max (pad = -inf is neutral)
    float m = -__builtin_inff();
#pragma unroll
    for (int k = 0; k < 8; ++k)
      m = fmaxf(m, fmaxf(fmaxf(r[k].x, r[k].y), fmaxf(r[k].z, r[k].w)));
#pragma unroll
    for (int off = 16; off > 0; off >>= 1)
      m = fmaxf(m, __shfl_xor(m, off, 32));

    // ---- pass 2: S = sum exp(x - m)   (exp(-inf - m) = 0 for pads)
    float4 e[8];
    float s = 0.0f;
#pragma unroll
    for (int k = 0; k < 8; ++k) {
      e[k].x = __expf(r[k].x - m);
      e[k].y = __expf(r[k].y - m);
      e[k].z = __expf(r[k].z - m);
      e[k].w = __expf(r[k].w - m);
      s += (e[k].x + e[k].y) + (e[k].z + e[k].w);
    }
#pragma unroll
    for (int off = 16; off > 0; off >>= 1)
      s += __shfl_xor(s, off, 32);

    // ---- pass 3: T = sum clip(e/S, 1e-7, 1)  (mask the 24 pad elements)
    const float invS = 1.0f / s;
    float t = 0.0f;
#pragma unroll
    for (int k = 0; k < 8; ++k) {
      const int i0 = (k * 32 + lane) * 4;
      t += (i0 + 0 < NUM_C) ? fminf(fmaxf(e[k].x * invS, 1e-7f), 1.0f) : 0.0f;
      t += (i0 + 1 < NUM_C) ? fminf(fmaxf(e[k].y * invS, 1e-7f), 1.0f) : 0.0f;
      t += (i0 + 2 < NUM_C) ? fminf(fmaxf(e[k].z * invS, 1e-7f), 1.0f) : 0.0f;
      t += (i0 + 3 < NUM_C) ? fminf(fmaxf(e[k].w * invS, 1e-7f), 1.0f) : 0.0f;
    }
#pragma unroll
    for (int off = 16; off > 0; off >>= 1)
      t += __shfl_xor(t, off, 32);

    // ---- target gather from LDS + per-row term
    const int   tgt = targets[row];
    const float xt  = buf[tgt];
    const float pt  = fminf(fmaxf(__expf(xt - m) * invS, 1e-7f), 1.0f);
    term = (m + __logf(s) - xt) + NEG_LOG_EPS * (t - pt);
  }

  if (lane == 0) sterm[wave] = term;
  __syncthreads();
  if (threadIdx.x == 0) {
    float acc = 0.0f;
#pragma unroll
    for (int w = 0; w < ROWS_PER_BLOCK; ++w) acc += sterm[w];
    partials[blockIdx.x] = acc;
  }
}

__global__ void __launch_bounds__(256)
sce_reduce_kernel(const float* __restrict__ partials, int n, float scale,
                  float* __restrict__ out) {
  __shared__ float red[256];
  float acc = 0.0f;
  for (int i = threadIdx.x; i < n; i += 256) acc += partials[i];
  red[threadIdx.x] = acc;
  __syncthreads();
  for (int s = 128; s > 0; s >>= 1) {
    if ((int)threadIdx.x < s) red[threadIdx.x] += red[threadIdx.x + s];
    __syncthreads();
  }
  if (threadIdx.x == 0) out[0] = red[0] * scale;
}

extern "C" void kernel_launch(void* const* d_in, const int* in_sizes, int n_in,
                              void* d_out, int out_size, void* d_ws, size_t ws_size,
                              hipStream_t stream) {
  const float* logits  = (const float*)d_in[0];
  const int*   targets = (const int*)d_in[1];
  const int n_rows = in_sizes[1];  // N = 65536

  float* partials = (float*)d_ws;  // n_blocks floats
  const int n_blocks = (n_rows + ROWS_PER_BLOCK - 1) / ROWS_PER_BLOCK;

  sce_row_kernel<<<n_blocks, 256, 0, stream>>>(logits, targets, partials, n_rows);
  sce_reduce_kernel<<<1, 256, 0, stream>>>(partials, n_blocks,
                                           1.0f / (float)n_rows, (float*)d_out);
}